// EdgeFeatureLayer_2001454760694
// MI455X (gfx1250) — compile-verified
//
#include <hip/hip_runtime.h>

// ---------------------------------------------------------------------------
// EdgeFeatureLayer for MI455X (gfx1250, wave32, WMMA)
//   out[b,i,j,:] = U_w @ e[b,i,j,:]  + U_b + (V_w @ x[b,i,:] + V_b)
//                                          + (V_w @ x[b,j,:] + V_b)
// Strategy: memory-bound fp32 GEMM (M=524288, K=128, N=128) on the fp32
// matrix pipe (v_wmma_f32_16x16x4_f32), single pass over e (256MB) and
// out (256MB); biases + row/col broadcasts fused in the epilogue via a
// precomputed S[b,n,:] = x@V_w^T + V_b + 0.5*U_b  (so S_i + S_j supplies
// U_b + V_b twice-added terms exactly).
// Roofline: 512MB mandatory traffic @ 23.3 TB/s ~= 22us floor; 17.2 GFLOP
// fp32 -> AI ~= 33.6 FLOP/B, near machine balance even on the fp32 matrix
// pipe, so fp32 WMMA keeps full reference precision at the memory roofline.
// ---------------------------------------------------------------------------

typedef __attribute__((ext_vector_type(2))) float v2f;
typedef __attribute__((ext_vector_type(4))) float v4f;
typedef __attribute__((ext_vector_type(8))) float v8f;

#define B_  8
#define N_  256
#define H_  128

// LDS row stride in dwords: 132 = 128 + 4 pad.  Fragment reads then hit
// banks (4*row + 4*k + {0..3}) mod 64 -> all 64 banks once, conflict-free.
#define LSTRIDE      132
#define UW_FLOATS    (H_ * LSTRIDE)        // U_w staged [o][k], 67584 B
#define ETILE_FLOATS (16 * LSTRIDE)        // one wave's 16 e-rows, 8448 B
#define WAVES_PER_WG 8
#define TILES_PER_WAVE 8
#define SMEM_FLOATS  (UW_FLOATS + WAVES_PER_WG * ETILE_FLOATS)
#define SMEM_BYTES   (SMEM_FLOATS * 4)     // 135168 B  (< 320KB/WGP, 2 WG/WGP)

// ---------------------------------------------------------------------------
// Kernel 1 (tiny, 0.1% of work): S[row][o] = sum_h x[row][h]*V_w[o][h]
//                                            + V_b[o] + 0.5f*U_b[o]
// ---------------------------------------------------------------------------
__global__ __launch_bounds__(H_) void vx_bias_kernel(
    const float* __restrict__ x,  const float* __restrict__ Vw,
    const float* __restrict__ Vb, const float* __restrict__ Ub,
    float* __restrict__ S)
{
  __shared__ float xs[H_];
  const int row = blockIdx.x;          // b*N + n  in [0, 2048)
  const int o   = threadIdx.x;         // [0, 128)
  xs[o] = x[row * H_ + o];
  __syncthreads();
  float acc = Vb[o] + 0.5f * Ub[o];
  const float* w = Vw + o * H_;
  #pragma unroll 8
  for (int h = 0; h < H_; ++h) acc = fmaf(xs[h], w[h], acc);
  S[row * H_ + o] = acc;
}

// ---------------------------------------------------------------------------
// Kernel 2 (main): streaming fp32 WMMA GEMM + fused epilogue.
// Each wave: 16-row x 128-col output tile; 32 K-steps x 8 O-tiles of
// v_wmma_f32_16x16x4_f32 with A/B fragments read as ds_load_b64 per the
// CDNA5 32-bit A/B VGPR layouts (lanes 0-15: K{0,1}; lanes 16-31: K{2,3}).
// ---------------------------------------------------------------------------
__global__ __launch_bounds__(256) void edge_gemm_kernel(
    const float* __restrict__ e,  const float* __restrict__ Uw,
    const float* __restrict__ S,  float* __restrict__ out)
{
  extern __shared__ float smem[];
  float* uw = smem;                                   // U_w, padded stride
  const int tid  = threadIdx.x;
  const int lane = tid & 31;
  const int wave = tid >> 5;
  float* etile = smem + UW_FLOATS + wave * ETILE_FLOATS;

  // Cooperative stage of U_w (row-major [o][h]) into padded LDS; coalesced.
  for (int idx = tid; idx < H_ * H_; idx += 256) {
    const int o = idx >> 7;
    const int k = idx & (H_ - 1);
    uw[o * LSTRIDE + k] = Uw[idx];
  }
  __syncthreads();

  const int n_ = lane & 15;   // column within 16-wide fragment
  const int hi = lane >> 4;   // half-wave select (K offset / M offset)

  const int waveGlobal = blockIdx.x * WAVES_PER_WG + wave;

  for (int it = 0; it < TILES_PER_WAVE; ++it) {
    const long tile    = (long)waveGlobal * TILES_PER_WAVE + it;
    const long rowBase = tile * 16;                  // global row m
    const int  bi = (int)(rowBase >> 8);             // b*N + i  (N_ == 256)
    const int  j0 = (int)(rowBase & (N_ - 1));       // tile's first j

    // Stage 16 e-rows (8 KB) into this wave's LDS slab: b128, coalesced.
    #pragma unroll 4
    for (int r = 0; r < 16; ++r) {
      const float* src = e + (rowBase + r) * H_ + lane * 4;
      *(v4f*)&etile[r * LSTRIDE + lane * 4] = *(const v4f*)src;
    }
    // Warm next tile in L2/WGP$ (gfx1250 global_prefetch_b8).
    if (it + 1 < TILES_PER_WAVE) {
      const float* nxt = e + (rowBase + 16) * H_;
      __builtin_prefetch(nxt + lane * 64, 0, 0);     // 32 lanes x 256B = 8KB
    }

    v8f acc[8] = {};                                 // 8 O-tiles x 8 VGPRs

    const int arow = n_ * LSTRIDE;                   // A-frag row base
    #pragma unroll 4
    for (int ks = 0; ks < 32; ++ks) {
      const int kb = ks * 4 + hi * 2;                // per-lane K base
      // A: e rows  (M = n_, K = kb, kb+1)
      const v2f a = *(const v2f*)&etile[arow + kb];
      #pragma unroll
      for (int ot = 0; ot < 8; ++ot) {
        // B[k][o] = U_w[o][k]:  (lane -> o column, VGPR -> K row)
        const v2f b = *(const v2f*)&uw[(ot * 16 + n_) * LSTRIDE + kb];
        acc[ot] = __builtin_amdgcn_wmma_f32_16x16x4_f32(
            /*neg_a=*/false, a, /*neg_b=*/false, b,
            /*c_mod=*/(short)0, acc[ot],
            /*reuse_a=*/false, /*reuse_b=*/false);
      }
    }

    // Epilogue: out[m][o] = acc + S[b,i][o] + S[b,j][o]; coalesced b32 stores
    // per the C/D layout: VGPR r -> M = r + hi*8, lane -> o = ot*16 + n_.
    const float* Si   = S + (long)bi * H_;
    const int    bRow = bi & ~(N_ - 1);              // b*N
    #pragma unroll
    for (int ot = 0; ot < 8; ++ot) {
      const int   o  = ot * 16 + n_;
      const float si = Si[o];
      #pragma unroll
      for (int r = 0; r < 8; ++r) {
        const int rr = r + hi * 8;
        const float sj = S[(long)(bRow + j0 + rr) * H_ + o];
        out[(rowBase + rr) * H_ + o] = acc[ot][r] + si + sj;
      }
    }
  }
}

// ---------------------------------------------------------------------------
extern "C" void kernel_launch(void* const* d_in, const int* in_sizes, int n_in,
                              void* d_out, int out_size, void* d_ws, size_t ws_size,
                              hipStream_t stream) {
  (void)in_sizes; (void)n_in; (void)out_size; (void)ws_size;
  const float* x  = (const float*)d_in[0];   // [B,N,H]
  const float* e  = (const float*)d_in[1];   // [B,N,N,H]
  const float* Uw = (const float*)d_in[2];   // [H,H]
  const float* Ub = (const float*)d_in[3];   // [H]
  const float* Vw = (const float*)d_in[4];   // [H,H]
  const float* Vb = (const float*)d_in[5];   // [H]
  float* out = (float*)d_out;                // [B,N,N,H]
  float* S   = (float*)d_ws;                 // [B*N, H] = 1 MiB scratch

  (void)hipFuncSetAttribute((const void*)edge_gemm_kernel,
                            hipFuncAttributeMaxDynamicSharedMemorySize,
                            SMEM_BYTES);

  vx_bias_kernel<<<B_ * N_, H_, 0, stream>>>(x, Vw, Vb, Ub, S);

  const int tilesTotal = (B_ * N_ * N_) / 16;                       // 32768
  const int grid = tilesTotal / (WAVES_PER_WG * TILES_PER_WAVE);    // 512
  edge_gemm_kernel<<<grid, 256, SMEM_BYTES, stream>>>(e, Uw, S, out);
}